// Renderer_51711406244136
// MI455X (gfx1250) — compile-verified
//
#include <hip/hip_runtime.h>
#include <math.h>

#define TILE   16
#define IMG_W  256
#define IMG_H  256
#define GSTRIDE 12      // floats per sorted gaussian record in workspace
#define MAXN   1024

typedef __attribute__((ext_vector_type(2))) float v2f;
typedef __attribute__((ext_vector_type(8))) float v8f;

// focal arrives as a 1-element array; reference uses python int 200.
// Accept either int32 or float32 encodings.
__device__ __forceinline__ float dev_scalar(const void* p) {
  int iv = *(const int*)p;
  if (iv > 0 && iv < (1 << 20)) return (float)iv;
  return *(const float*)p;
}

// ---------------------------------------------------------------------------
// Kernel 1: per-gaussian geometry + stable depth sort (single block, N threads)
// Writes sorted records: [i00, s01, i11, mx, my, O, r, g, b, bbox, pad, pad]
// ---------------------------------------------------------------------------
__global__ void geom_sort_kernel(const float* __restrict__ mu,
                                 const float* __restrict__ scales,
                                 const float* __restrict__ quats,
                                 const float* __restrict__ cols,
                                 const float* __restrict__ opcs,
                                 const float* __restrict__ viewmat,
                                 const void* __restrict__ focal_p,
                                 float* __restrict__ gws, int N)
{
  __shared__ float s_depth[MAXN];
  int i = threadIdx.x;
  if (i >= N) return;

  float fx = dev_scalar(focal_p), fy = fx;
  const float W = (float)IMG_W, H = (float)IMG_H;
  const float cx = W * 0.5f, cy = H * 0.5f;
  const float n_ = 0.01f, f_ = 10.0f;

  float V[16];
#pragma unroll
  for (int k = 0; k < 16; ++k) V[k] = viewmat[k];

  float mx3 = mu[i*3+0], my3 = mu[i*3+1], mz3 = mu[i*3+2];
  float t0 = V[0]*mx3 + V[1]*my3 + V[2]*mz3  + V[3];
  float t1 = V[4]*mx3 + V[5]*my3 + V[6]*mz3  + V[7];
  float t2 = V[8]*mx3 + V[9]*my3 + V[10]*mz3 + V[11];
  float t3 = V[12]*mx3 + V[13]*my3 + V[14]*mz3 + V[15];

  float tp0 = 2.0f*fx/W * t0;
  float tp1 = 2.0f*fy/H * t1;
  float depth = (f_+n_)/(f_-n_) * t2 + (-2.0f*f_*n_/(f_-n_)) * t3;
  float mx = W * tp0 / t2 / 2.0f + cx;
  float my = H * tp1 / t2 / 2.0f + cy;

  // rotation from quat (replicating the reference formulas verbatim,
  // including the 1-2(x^2 - z^2) term)
  float qx = quats[i*4+0], qy = quats[i*4+1], qz = quats[i*4+2], qw = quats[i*4+3];
  float R00 = 1.f-2.f*(qy*qy+qz*qz), R01 = 2.f*(qx*qy-qw*qz), R02 = 2.f*(qx*qz+qw*qy);
  float R10 = 2.f*(qx*qy+qw*qz), R11 = 1.f-2.f*(qx*qx-qz*qz), R12 = 2.f*(qy*qz-qw*qx);
  float R20 = 2.f*(qx*qz+qw*qy), R21 = 2.f*(qy*qz+qw*qx), R22 = 1.f-2.f*(qx*qx+qy*qy);

  float s0 = scales[i*3+0], s1 = scales[i*3+1], s2 = scales[i*3+2];
  float RS00=R00*s0, RS01=R01*s1, RS02=R02*s2;
  float RS10=R10*s0, RS11=R11*s1, RS12=R12*s2;
  float RS20=R20*s0, RS21=R21*s1, RS22=R22*s2;

  // Sigma = RS RS^T (symmetric)
  float S00 = RS00*RS00 + RS01*RS01 + RS02*RS02;
  float S01 = RS00*RS10 + RS01*RS11 + RS02*RS12;
  float S02 = RS00*RS20 + RS01*RS21 + RS02*RS22;
  float S11 = RS10*RS10 + RS11*RS11 + RS12*RS12;
  float S12 = RS10*RS20 + RS11*RS21 + RS12*RS22;
  float S22 = RS20*RS20 + RS21*RS21 + RS22*RS22;

  float tz = t2;
  float J00 = fx/tz,  J02 = -fx*t0/(tz*tz);
  float J11 = fy/tz,  J12 = -fy*t1/(tz*tz);

  // M = J @ Rcw  (Rcw = viewmat[:3,:3]); J01=J10=0
  float M00 = J00*V[0] + J02*V[8];
  float M01 = J00*V[1] + J02*V[9];
  float M02 = J00*V[2] + J02*V[10];
  float M10 = J11*V[4] + J12*V[8];
  float M11 = J11*V[5] + J12*V[9];
  float M12 = J11*V[6] + J12*V[10];

  // tmp = M * Sigma (2x3), cov = tmp * M^T (2x2)
  float T00 = M00*S00 + M01*S01 + M02*S02;
  float T01 = M00*S01 + M01*S11 + M02*S12;
  float T02 = M00*S02 + M01*S12 + M02*S22;
  float T10 = M10*S00 + M11*S01 + M12*S02;
  float T11 = M10*S01 + M11*S11 + M12*S12;
  float T12 = M10*S02 + M11*S12 + M12*S22;
  float a    = T00*M00 + T01*M01 + T02*M02;
  float b    = T00*M10 + T01*M11 + T02*M12;   // cov[0,1]
  float c10v = T10*M00 + T11*M01 + T12*M02;   // cov[1,0]
  float d    = T10*M10 + T11*M11 + T12*M12;

  float Aev = sqrtf(a*a - 2.f*a*d + 4.f*b*b + d*d);
  float Bev = a + d;
  float l1 = 0.5f*(Aev + Bev), l2 = 0.5f*(Bev - Aev);
  float nanv = __int_as_float(0x7fc00000);
  float r1 = (l1 > 0.f) ? 3.f*sqrtf(l1) : nanv;
  float r2 = (l2 > 0.f) ? 3.f*sqrtf(l2) : nanv;
  float theta = (b != 0.f) ? atan2f(l1 - a, b) : ((a >= d) ? 1.5707963267948966f : 0.f);
  float cth = cosf(theta), sth = sinf(theta);

  float minx, maxx, miny, maxy;
  if (isnan(r1) || isnan(r2)) {
    // any NaN radius propagates through rotation -> bbs replaced by -1
    minx = maxx = miny = maxy = -1.f;
  } else {
    minx = miny = 3.4e38f; maxx = maxy = -3.4e38f;
    const float cdx[4] = {-1.f, 1.f, 1.f, -1.f};
    const float cdy[4] = { 1.f, 1.f,-1.f, -1.f};
#pragma unroll
    for (int k = 0; k < 4; ++k) {
      float dx = cdx[k]*r1, dy = cdy[k]*r2;
      float bx = cth*dx - sth*dy + mx;
      float by = sth*dx + cth*dy + my;
      minx = fminf(minx, bx); maxx = fmaxf(maxx, bx);
      miny = fminf(miny, by); maxy = fmaxf(maxy, by);
    }
  }
  const float WtF = (float)(IMG_W/TILE - 1), HtF = (float)(IMG_H/TILE - 1);
  int xmin = (int)fminf(fmaxf(floorf(minx/(float)TILE), 0.f), WtF);
  int xmax = (int)fminf(fmaxf(ceilf (maxx/(float)TILE), 0.f), WtF);
  int ymin = (int)fminf(fmaxf(floorf(miny/(float)TILE), 0.f), HtF);
  int ymax = (int)fminf(fmaxf(ceilf (maxy/(float)TILE), 0.f), HtF);
  int bbox = xmin | (xmax << 8) | (ymin << 16) | (ymax << 24);

  const float e = 1e-6f;
  float det = (a+e)*(d+e) - b*c10v;
  float i00 = (d+e)/det;
  float i01 = -b/det, i10 = -c10v/det;
  float i11 = (a+e)/det;
  float s01 = i01 + i10;

  float ov = opcs[i];
  float sg = 1.f/(1.f + expf(-ov));
  float O  = 1.f/(1.f + expf(-sg));
  float cr = fminf(fmaxf(cols[i*3+0], 1e-4f), 1.f);
  float cg = fminf(fmaxf(cols[i*3+1], 1e-4f), 1.f);
  float cb = fminf(fmaxf(cols[i*3+2], 1e-4f), 1.f);

  // stable ascending rank (matches jnp.argsort)
  s_depth[i] = depth;
  __syncthreads();
  int rank = 0;
  for (int j = 0; j < N; ++j) {
    float dj = s_depth[j];
    rank += ((dj < depth) || (dj == depth && j < i)) ? 1 : 0;
  }
  float* o = gws + rank * GSTRIDE;
  o[0]=i00; o[1]=s01; o[2]=i11; o[3]=mx; o[4]=my;
  o[5]=O;   o[6]=cr;  o[7]=cg;  o[8]=cb; o[9]=__int_as_float(bbox);
}

// ---------------------------------------------------------------------------
// Kernel 2: per-tile render. One block = one 16x16 pixel tile, 256 threads.
// quad[p,g] evaluated as rank-6 GEMM via V_WMMA_F32_16X16X4_F32.
// ---------------------------------------------------------------------------
__global__ void render_kernel(const float* __restrict__ gws,
                              float* __restrict__ out, int N)
{
  __shared__ int   s_list[MAXN];
  __shared__ int   s_cnt;
  __shared__ float s_q[256 * 17];   // quad values, stride 17 to dodge bank conflicts
  __shared__ float s_gp[16 * 6];    // i00, s01, i11, mxl, myl per staged gaussian
  __shared__ float s_gc[16 * 4];    // r, g, b, O per staged gaussian

  int tid  = threadIdx.x;
  int lane = tid & 31;
  int wv   = tid >> 5;
  int tx   = blockIdx.x, ty = blockIdx.y;
  float ox = (float)(tx*TILE) + 7.5f;   // tile-centered local coordinates
  float oy = (float)(ty*TILE) + 7.5f;

  // ---- order-preserving coverage compaction (wave 0, ballot-based) ----
  if (wv == 0) {
    int cnt = 0;
    for (int base = 0; base < N; base += 32) {
      int s = base + lane;
      bool pred = false;
      if (s < N) {
        int bbox = __float_as_int(gws[s*GSTRIDE + 9]);
        int xmin =  bbox        & 255;
        int xmax = (bbox >> 8)  & 255;
        int ymin = (bbox >> 16) & 255;
        int ymax = (bbox >> 24) & 255;
        pred = (tx >= xmin) && (tx <= xmax) && (ty >= ymin) && (ty <= ymax);
      }
      unsigned m = (unsigned)__ballot(pred);   // wave32: low 32 bits
      int ofs = __popc(m & ((1u << lane) - 1u));
      if (pred && (cnt + ofs) < MAXN) s_list[cnt + ofs] = s;
      cnt += __popc(m);
    }
    if (lane == 0) s_cnt = cnt;
  }
  __syncthreads();
  int cnt = s_cnt;

  int px = tid & 15, py = tid >> 4;
  int oidx = ((ty*TILE + py) * IMG_W + (tx*TILE + px)) * 3;
  if (cnt == 0) {               // tile untouched by any bbox -> zeros (uniform exit)
    out[oidx+0] = 0.f; out[oidx+1] = 0.f; out[oidx+2] = 0.f;
    return;
  }

  // ---- loop-invariant A fragments: pixel features [x^2, xy | y^2, x] ; [y, 1 | 0, 0]
  // A layout (16x4 f32): lanes 0-15 row M=lane, VGPR0=K0 VGPR1=K1; lanes 16-31 K2,K3.
  bool hi = (lane >= 16);
  v2f A1[2], A2[2];
#pragma unroll
  for (int t = 0; t < 2; ++t) {
    int m = 2*wv + t;                 // wave covers pixel rows [32w, 32w+32)
    int p = m*16 + (lane & 15);       // A row = pixel index in tile
    float xl = (float)(p & 15) - 7.5f;
    float yl = (float)(p >> 4) - 7.5f;
    if (hi) { A1[t].x = yl*yl; A1[t].y = xl;    A2[t].x = 0.f; A2[t].y = 0.f; }
    else    { A1[t].x = xl*xl; A1[t].y = xl*yl; A2[t].x = yl;  A2[t].y = 1.f; }
  }

  float Tcur = 1.f, accR = 0.f, accG = 0.f, accB = 0.f;

  for (int b0 = 0; b0 < cnt; b0 += 16) {
    // ---- stage 16 gaussians into LDS (tile-local centers); lanes 16..31 of
    // wave 0 software-prefetch the NEXT batch's gathered records into cache
    // (global_prefetch_b8) to hide the indexed-gather latency.
    if (tid < 16) {
      int slot = b0 + tid;
      if (slot < cnt) {
        const float* g = gws + s_list[slot]*GSTRIDE;
        s_gp[tid*6+0] = g[0]; s_gp[tid*6+1] = g[1]; s_gp[tid*6+2] = g[2];
        s_gp[tid*6+3] = g[3] - ox; s_gp[tid*6+4] = g[4] - oy;
        s_gc[tid*4+0] = g[6]; s_gc[tid*4+1] = g[7];
        s_gc[tid*4+2] = g[8]; s_gc[tid*4+3] = g[5];
      } else {
        s_gp[tid*6+0]=0.f; s_gp[tid*6+1]=0.f; s_gp[tid*6+2]=0.f;
        s_gp[tid*6+3]=0.f; s_gp[tid*6+4]=0.f;
        s_gc[tid*4+0]=0.f; s_gc[tid*4+1]=0.f; s_gc[tid*4+2]=0.f; s_gc[tid*4+3]=0.f;
      }
    } else if (tid < 32) {
      int slot = b0 + tid;            // == b0 + 16 + (tid-16): next batch
      if (slot < cnt) {
        const float* g = gws + s_list[slot]*GSTRIDE;
        __builtin_prefetch(g, 0, 3);          // first half of record
        __builtin_prefetch(g + 8, 0, 3);      // tail (may cross a line)
      }
    }
    __syncthreads();

    // ---- B fragments: per-gaussian bilinear coefficients (rank-6, K padded to 8)
    // B layout (4x16 f32): lanes 0-15 col N=lane, VGPR0=K0 VGPR1=K1; lanes 16-31 K2,K3.
    {
      int g = lane & 15;
      float i00 = s_gp[g*6+0], s01 = s_gp[g*6+1], i11 = s_gp[g*6+2];
      float mxl = s_gp[g*6+3], myl = s_gp[g*6+4];
      float P  = -(2.f*i00*mxl + s01*myl);
      float Q  = -(s01*mxl + 2.f*i11*myl);
      float Rv = i00*mxl*mxl + s01*mxl*myl + i11*myl*myl;
      v2f B1, B2;
      if (hi) { B1.x = i11; B1.y = P;   B2.x = 0.f; B2.y = 0.f; }
      else    { B1.x = i00; B1.y = s01; B2.x = Q;   B2.y = Rv;  }

#pragma unroll
      for (int t = 0; t < 2; ++t) {
        v8f c = {};
        c = __builtin_amdgcn_wmma_f32_16x16x4_f32(false, A1[t], false, B1,
                                                  (short)0, c, false, false);
        c = __builtin_amdgcn_wmma_f32_16x16x4_f32(false, A2[t], false, B2,
                                                  (short)0, c, false, false);
        // C/D layout: VGPR v holds M=v (lanes 0-15) / M=v+8 (lanes 16-31), N=lane%16
        int p0 = (2*wv + t)*16 + (hi ? 8 : 0);
#pragma unroll
        for (int v = 0; v < 8; ++v) s_q[(p0 + v)*17 + g] = c[v];
      }
    }
    __syncthreads();

    // ---- per-pixel front-to-back compositing over this batch ----
    // __expf -> single v_exp_f32 on the transcendental pipe (co-executes with
    // VALU on CDNA5); saturation behavior (0 / +inf / NaN) matches expf.
    int jn = min(16, cnt - b0);
    for (int j = 0; j < jn; ++j) {
      float q  = s_q[tid*17 + j];
      float pg = __expf(-0.5f * q);
      if (isnan(pg)) pg = 0.f;                      // jnp.nan_to_num
      else if (isinf(pg)) pg = 3.402823466e38f;
      float alpha = s_gc[j*4+3] * pg;
      if (alpha < 0.001f) alpha = 0.f;
      alpha = fminf(alpha, 0.99f);
      float w = alpha * Tcur;
      accR += w * s_gc[j*4+0];
      accG += w * s_gc[j*4+1];
      accB += w * s_gc[j*4+2];
      Tcur *= (1.f - alpha);
    }
    __syncthreads();
  }

  out[oidx+0] = fminf(fmaxf(accR, 1e-4f), 1.f);
  out[oidx+1] = fminf(fmaxf(accG, 1e-4f), 1.f);
  out[oidx+2] = fminf(fmaxf(accB, 1e-4f), 1.f);
}

extern "C" void kernel_launch(void* const* d_in, const int* in_sizes, int n_in,
                              void* d_out, int out_size, void* d_ws, size_t ws_size,
                              hipStream_t stream)
{
  const float* mu      = (const float*)d_in[0];
  const float* scales  = (const float*)d_in[1];
  const float* quats   = (const float*)d_in[2];
  const float* cols    = (const float*)d_in[3];
  const float* opcs    = (const float*)d_in[4];
  const float* viewmat = (const float*)d_in[5];
  const void*  focal   = d_in[7];
  int N = in_sizes[4];              // opcs has N elements
  if (N > MAXN) N = MAXN;
  float* gws = (float*)d_ws;        // needs N*GSTRIDE*4 bytes (24 KB for N=512)

  geom_sort_kernel<<<1, N, 0, stream>>>(mu, scales, quats, cols, opcs,
                                        viewmat, focal, gws, N);
  render_kernel<<<dim3(IMG_W/TILE, IMG_H/TILE), 256, 0, stream>>>(
      gws, (float*)d_out, N);
}